// GCNEncoder_44487271252060
// MI455X (gfx1250) — compile-verified
//
#include <hip/hip_runtime.h>
#include <hip/hip_bf16.h>

typedef __attribute__((ext_vector_type(2))) float v2f;
typedef __attribute__((ext_vector_type(8))) float v8f;

#define N_NODES 50000
#define N_EDGES 800000
#define DIM 128

// ---------------- zero fill ----------------
__global__ void zero_u32_kernel(unsigned* __restrict__ p, int n) {
    int i = blockIdx.x * blockDim.x + threadIdx.x;
    if (i < n) p[i] = 0u;
}

// ---------------- degree histograms ----------------
__global__ void degree_kernel(const int* __restrict__ src, const int* __restrict__ dst,
                              int* __restrict__ deg_out, int* __restrict__ deg_in, int ne) {
    int e = blockIdx.x * blockDim.x + threadIdx.x;
    if (e < ne) {
        atomicAdd(&deg_out[src[e]], 1);
        atomicAdd(&deg_in[dst[e]], 1);
    }
}

// ---------------- norms: clip(deg,1)^-0.5 ----------------
__global__ void norm_kernel(const int* __restrict__ deg_out, const int* __restrict__ deg_in,
                            float* __restrict__ norm_src, float* __restrict__ norm_dst, int n) {
    int i = blockIdx.x * blockDim.x + threadIdx.x;
    if (i < n) {
        int dov = deg_out[i]; if (dov < 1) dov = 1;
        int din = deg_in[i];  if (din < 1) din = 1;
        norm_src[i] = rsqrtf((float)dov);
        norm_dst[i] = rsqrtf((float)din);
    }
}

// ---------------- H = (x * norm_src[:,None]) @ W  via fp32 WMMA 16x16x4 ----------------
// One wave per 16-row strip; wave computes full 16x128 output (8 N-tiles).
// A 16x4 f32 frag: lane l holds m = l&15, k-pair base kk = 2*(l>>4), regs {k+kk, k+kk+1}.
// B 4x16 f32 frag: lane l holds n = l&15, rows {k+kk, k+kk+1} in the two regs.
// C 16x16 f32: reg r holds row r + 8*(l>>4), col l&15.
__global__ __launch_bounds__(256) void gemm_wmma_kernel(
    const float* __restrict__ x, const float* __restrict__ norm_src,
    const float* __restrict__ W, float* __restrict__ H) {
    int wave = (int)((blockIdx.x * blockDim.x + threadIdx.x) >> 5);
    int lane = threadIdx.x & 31;
    const int n_tiles = N_NODES / 16;   // 3125, exact
    if (wave >= n_tiles) return;        // wave-uniform branch: EXEC all-ones inside

    const int row0 = wave * 16;
    const int m    = lane & 15;         // A row / B col / C col index for this lane
    const int kk   = (lane >> 4) * 2;   // k sub-pair selector
    const int mh   = (lane >> 4) * 8;   // C row-half selector

    const float nrm = norm_src[row0 + m];
    const float* xrow = x + (size_t)(row0 + m) * DIM + kk;

    v8f acc[8] = {};

    for (int k = 0; k < DIM; k += 4) {
        float2 xv = *(const float2*)(xrow + k);
        v2f a = { xv.x * nrm, xv.y * nrm };
        const float* wb = W + (size_t)(k + kk) * DIM + m;
#pragma unroll
        for (int t = 0; t < 8; ++t) {
            v2f b = { wb[t * 16], wb[t * 16 + DIM] };
            acc[t] = __builtin_amdgcn_wmma_f32_16x16x4_f32(
                false, a, false, b, (short)0, acc[t], false, false);
        }
    }

    float* hrow = H + (size_t)row0 * DIM + m;   // col = m
#pragma unroll
    for (int t = 0; t < 8; ++t) {
#pragma unroll
        for (int r = 0; r < 8; ++r) {
            hrow[(size_t)(r + mh) * DIM + t * 16] = acc[t][r];
        }
    }
}

// ---------------- edge scatter: out[dst] += H[src]  (one wave per edge) ----------------
__global__ __launch_bounds__(256) void scatter_kernel(
    const float* __restrict__ H, const int* __restrict__ src, const int* __restrict__ dst,
    float* __restrict__ out, int ne) {
    int wave = (int)((blockIdx.x * blockDim.x + threadIdx.x) >> 5);
    int lane = threadIdx.x & 31;
    if (wave >= ne) return;
    int s = src[wave];
    int d = dst[wave];
    const float4 v = *(const float4*)(H + (size_t)s * DIM + lane * 4);
    float* o = out + (size_t)d * DIM + lane * 4;
    atomicAdd(o + 0, v.x);
    atomicAdd(o + 1, v.y);
    atomicAdd(o + 2, v.z);
    atomicAdd(o + 3, v.w);
}

// ---------------- finish: out = out * norm_dst[:,None] + b ----------------
__global__ void finish_kernel(float* __restrict__ out, const float* __restrict__ norm_dst,
                              const float* __restrict__ b, int n) {
    int idx = blockIdx.x * blockDim.x + threadIdx.x;
    if (idx < n) {
        int i = idx >> 7;       // /128
        int j = idx & (DIM - 1);
        out[idx] = out[idx] * norm_dst[i] + b[j];
    }
}

extern "C" void kernel_launch(void* const* d_in, const int* in_sizes, int n_in,
                              void* d_out, int out_size, void* d_ws, size_t ws_size,
                              hipStream_t stream) {
    const float* x   = (const float*)d_in[0];
    const int*   src = (const int*)d_in[1];
    const int*   dst = (const int*)d_in[2];
    const float* W   = (const float*)d_in[3];
    const float* b   = (const float*)d_in[4];
    float* out = (float*)d_out;

    // workspace layout
    int*   deg_out  = (int*)d_ws;
    int*   deg_in   = deg_out + N_NODES;
    float* norm_src = (float*)(deg_in + N_NODES);
    float* norm_dst = norm_src + N_NODES;
    float* H        = norm_dst + N_NODES;        // 50000*128 floats, 16B-aligned

    const int TB = 256;

    // 1) zero output accumulator and degree arrays
    zero_u32_kernel<<<(N_NODES * DIM + TB - 1) / TB, TB, 0, stream>>>((unsigned*)out, N_NODES * DIM);
    zero_u32_kernel<<<(2 * N_NODES + TB - 1) / TB, TB, 0, stream>>>((unsigned*)deg_out, 2 * N_NODES);

    // 2) degrees
    degree_kernel<<<(N_EDGES + TB - 1) / TB, TB, 0, stream>>>(src, dst, deg_out, deg_in, N_EDGES);

    // 3) norms
    norm_kernel<<<(N_NODES + TB - 1) / TB, TB, 0, stream>>>(deg_out, deg_in, norm_src, norm_dst, N_NODES);

    // 4) H = (x * norm_src) @ W   (fp32 WMMA)
    {
        int waves  = N_NODES / 16;                 // 3125
        int blocks = (waves * 32 + TB - 1) / TB;   // 391
        gemm_wmma_kernel<<<blocks, TB, 0, stream>>>(x, norm_src, W, H);
    }

    // 5) scatter-add over edges into out
    {
        long long threads = (long long)N_EDGES * 32;
        int blocks = (int)((threads + TB - 1) / TB);  // 100000
        scatter_kernel<<<blocks, TB, 0, stream>>>(H, src, dst, out, N_EDGES);
    }

    // 6) out = out * norm_dst + b
    finish_kernel<<<(N_NODES * DIM + TB - 1) / TB, TB, 0, stream>>>(out, norm_dst, b, N_NODES * DIM);
}